// loss_v2_51840255263198
// MI455X (gfx1250) — compile-verified
//
#include <hip/hip_runtime.h>

// ---------------------------------------------------------------------------
// YOLO-v2 loss for MI455X (gfx1250, wave32).
// Dims fixed by the reference: nA=5, nC=2, T=50, nH=nW=96, 8 channels/anchor.
// nB derived from in_sizes[1].
// ---------------------------------------------------------------------------

#define NA 5
#define NC 2
#define TT 50
#define NH 96
#define NW 96
#define NCH 8                      // 5 + nC + 1
#define CELLS_PER_B (NA * NH * NW) // 46080 = 180 * 256
#define SIL_THRESH 0.6f
#define OBJECT_SCALE 5.0f

__constant__ float c_aw[NA] = {1.3221f, 3.19275f, 5.05587f, 9.47112f, 11.2364f};
__constant__ float c_ah[NA] = {1.73145f, 4.00944f, 8.09892f, 4.84053f, 10.0071f};

typedef __attribute__((ext_vector_type(2))) float v2f;
typedef __attribute__((ext_vector_type(8))) float v8f;

__device__ __forceinline__ float sigmoidf(float v) {
  return 1.0f / (1.0f + __expf(-v));
}

__device__ __forceinline__ float iou_cwh(float x1, float y1, float w1, float h1,
                                         float x2, float y2, float w2, float h2) {
  float uw = fmaxf(x1 + w1 * 0.5f, x2 + w2 * 0.5f) - fminf(x1 - w1 * 0.5f, x2 - w2 * 0.5f);
  float uh = fmaxf(y1 + h1 * 0.5f, y2 + h2 * 0.5f) - fminf(y1 - h1 * 0.5f, y2 - h2 * 0.5f);
  float cw = w1 + w2 - uw;
  float ch = h1 + h2 - uh;
  float carea = (cw <= 0.0f || ch <= 0.0f) ? 0.0f : cw * ch;
  return carea / (w1 * h1 + w2 * h2 - carea);
}

// ---------------------------------------------------------------------------
// Kernel 0: per-(b,t) ground-truth preprocessing. One thread per batch so the
// sequential cumprod-validity is trivial. Writes 12 floats per (b,t):
// [gx, gy, gw, gh, tx, ty, tw, th, tcls, lr, best_n, valid]
// ---------------------------------------------------------------------------
__global__ void yolo_prep_kernel(const float* __restrict__ target,
                                 float* __restrict__ ws, int nB) {
  int b = blockIdx.x * blockDim.x + threadIdx.x;
  if (b >= nB) return;
  bool valid = true;
  for (int t = 0; t < TT; ++t) {
    const float* tg = target + (size_t)(b * TT + t) * 6;
    float tcls = tg[0];
    float gx = tg[1] * (float)NW;
    float gy = tg[2] * (float)NH;
    float gw = tg[3] * (float)NW;
    float gh = tg[4] * (float)NH;
    float lr = tg[5];
    valid = valid && (tg[1] != 0.0f);
    // best anchor: argmax over shape-IoU of (0,0,aw,ah) vs (0,0,gw,gh)
    int bestn = 0;
    float best = -1.0f;
    for (int n = 0; n < NA; ++n) {
      float inter = fminf(c_aw[n], gw) * fminf(c_ah[n], gh);
      float u = c_aw[n] * c_ah[n] + gw * gh - inter;
      float iou = inter / u;
      if (iou > best) { best = iou; bestn = n; }
    }
    int gi = (int)gx;
    int gj = (int)gy;
    float* o = ws + (size_t)(b * TT + t) * 12;
    o[0] = gx;
    o[1] = gy;
    o[2] = gw;
    o[3] = gh;
    o[4] = gx - (float)gi;
    o[5] = gy - (float)gj;
    o[6] = __logf(gw / c_aw[bestn]);
    o[7] = __logf(gh / c_ah[bestn]);
    o[8] = tcls;
    o[9] = lr;
    o[10] = (float)bestn;
    o[11] = valid ? 1.0f : 0.0f;
  }
}

// ---------------------------------------------------------------------------
// Kernel 1: fused per-cell loss. grid = (180, nB), block = 256 (8 waves).
// Each thread owns one (a,j,i) cell of batch blockIdx.y. GT records are staged
// in LDS. Scatter semantics ("last valid t wins") reproduced by scanning t
// ascending and keeping the final match. Wave reduction uses
// V_WMMA_F32_16X16X4_F32 (full fp32) + one shfl_xor(16).
// ---------------------------------------------------------------------------
__global__ void __launch_bounds__(256)
yolo_loss_kernel(const float* __restrict__ out, const float* __restrict__ gtws,
                 float* __restrict__ loss, float inv_nB) {
  __shared__ float sgt[TT * 12];
  __shared__ float swave[8];

  const int tid = threadIdx.x;
  const int b = blockIdx.y;
  const int cell = blockIdx.x * 256 + tid;

  for (int k = tid; k < TT * 12; k += 256) sgt[k] = gtws[(size_t)b * TT * 12 + k];
  __syncthreads();

  float partial = 0.0f;
  if (cell < CELLS_PER_B) {
    const int a = cell / (NH * NW);
    const int rem = cell - a * (NH * NW);
    const int j = rem / NW;
    const int i = rem - j * NW;

    const size_t base = ((size_t)b * (NA * NCH) + (size_t)a * NCH) * (NH * NW) +
                        (size_t)j * NW + i;
    float c[NCH];
#pragma unroll
    for (int ch = 0; ch < NCH; ++ch) c[ch] = out[base + (size_t)ch * (NH * NW)];

    const float x = sigmoidf(c[0]);
    const float y = sigmoidf(c[1]);
    const float conf = sigmoidf(c[4]);
    const float conf_lr = sigmoidf(c[7]);
    const float px = x + (float)i;
    const float py = y + (float)j;
    const float pw = __expf(c[2]) * c_aw[a];
    const float ph = __expf(c[3]) * c_ah[a];

    float curmax = 0.0f;
    int thit = -1;
    for (int t = 0; t < TT; ++t) {
      const float* g = &sgt[t * 12];
      if (g[11] == 0.0f) break; // validity is a prefix (cumprod)
      float iou = iou_cwh(px, py, pw, ph, g[0], g[1], g[2], g[3]);
      curmax = fmaxf(curmax, iou);
      if ((int)g[10] == a && (int)g[0] == i && (int)g[1] == j) thit = t;
    }

    if (thit >= 0) {
      const float* g = &sgt[thit * 12];
      // tconf = IoU(gt, pred at scattered cell) == IoU(gt, this cell's box)
      const float tconf = iou_cwh(g[0], g[1], g[2], g[3], px, py, pw, ph);
      const float dx = x - g[4];
      const float dy = y - g[5];
      const float dw = c[2] - g[6];
      const float dh = c[3] - g[7];
      partial += 0.5f * (dx * dx + dy * dy + dw * dw + dh * dh); // coord terms
      const float dc = conf - tconf;
      partial += 0.5f * OBJECT_SCALE * dc * dc;                  // object conf
      // class cross-entropy over 2 logits
      const float l0 = c[5], l1 = c[6];
      const float m = fmaxf(l0, l1);
      const float lse = m + __logf(__expf(l0 - m) + __expf(l1 - m));
      const float lt = ((int)g[8] == 0) ? l0 : l1;
      partial += (lse - lt);
      const float dl = conf_lr - g[9];
      partial += 0.25f * dl * dl;                                // 0.5 * loss_conf_lr
    } else {
      const float cm0 = (curmax > SIL_THRESH) ? 0.0f : 1.0f;     // NOOBJECT_SCALE=1
      partial += 0.5f * cm0 * conf * conf;                       // tconf == 0 here
    }
  }

  // ---- wave32 sum via the matrix pipe (full fp32): D = A(16x4) * ones(4x16)
  // A vgpr0 lane m      -> A[m][0] = partial[lane m]
  // A vgpr0 lane m+16   -> A[m][2] = partial[lane m+16]; vgpr1 zeroed.
  // D[m][n] = partial[m] + partial[m+16]. Lane l holds D[0..7][l] in 8 VGPRs,
  // lane l+16 holds D[8..15][l] -> total = s(l) + s(l^16).
  v2f amat;
  amat[0] = partial;
  amat[1] = 0.0f;
  v2f bones;
  bones[0] = 1.0f;
  bones[1] = 1.0f;
  v8f cacc = {0.f, 0.f, 0.f, 0.f, 0.f, 0.f, 0.f, 0.f};
  v8f d = __builtin_amdgcn_wmma_f32_16x16x4_f32(
      /*neg_a=*/false, amat, /*neg_b=*/false, bones,
      /*c_mod=*/(short)0, cacc, /*reuse_a=*/false, /*reuse_b=*/false);
  float s = d[0] + d[1] + d[2] + d[3] + d[4] + d[5] + d[6] + d[7];
  s += __shfl_xor(s, 16, 32);

  const int lane = tid & 31;
  const int wave = tid >> 5;
  if (lane == 0) swave[wave] = s;
  __syncthreads();
  if (tid == 0) {
    float bs = 0.0f;
#pragma unroll
    for (int w = 0; w < 8; ++w) bs += swave[w];
    atomicAdd(loss, bs * inv_nB);
  }
}

// ---------------------------------------------------------------------------
extern "C" void kernel_launch(void* const* d_in, const int* in_sizes, int n_in,
                              void* d_out, int out_size, void* d_ws, size_t ws_size,
                              hipStream_t stream) {
  const float* output = (const float*)d_in[0];
  const float* target = (const float*)d_in[1];
  float* loss = (float*)d_out;
  float* ws = (float*)d_ws; // needs nB*T*12 floats (= 38.4 KB at nB=16)

  const int nB = in_sizes[1] / (TT * 6);

  hipMemsetAsync(d_out, 0, sizeof(float), stream);

  yolo_prep_kernel<<<1, 32, 0, stream>>>(target, ws, nB);

  dim3 grid(CELLS_PER_B / 256, nB); // 180 x nB, exact -> EXEC all-ones at WMMA
  yolo_loss_kernel<<<grid, 256, 0, stream>>>(output, ws, loss, 1.0f / (float)nB);
}